// hnet_structure_MLP_55465207661028
// MI455X (gfx1250) — compile-verified
//
#include <hip/hip_runtime.h>

typedef __attribute__((ext_vector_type(2))) float v2f;
typedef __attribute__((ext_vector_type(8))) float v8f;

struct Ptrs { const float* p[35]; };

// Per-MLP metadata
__device__ __constant__ int d_npe[7]     = {8,4,2,2,4,4,2};       // batch rows
__device__ __constant__ int d_insz[7]    = {4,3,2,2,3,3,2};       // bits + SoW
__device__ __constant__ int d_h[7]       = {128,78,45,45,78,78,45};
__device__ __constant__ int d_nhid[7]    = {3,2,1,1,1,1,1};
__device__ __constant__ int d_firstW[7]  = {1,9,15,19,23,27,31};  // d_in index of first hidden W
__device__ __constant__ int d_finalW[7]  = {7,13,17,21,25,29,33}; // d_in index of final W
__device__ __constant__ int d_osz[7]     = {4096,2048,1024,1024,2048,2048,1024};
__device__ __constant__ int d_startIdx[7]= {0,8,12,14,16,20,24};  // first original output index
// cumulative 16-wide output tiles per MLP: {256,128,64,64,128,128,64}
__device__ __constant__ int d_tilecum[8] = {0,256,384,448,512,640,768,832};
// d_out float offset for each ORIGINAL output index (derived from OUTPUT_ORDER)
__device__ __constant__ int d_outoff[26] = {
  26624,22528,34816,30720,43008,38912,51200,47104,   // idx 0..7  (MLP0, 4096)
  57344,55296,61440,59392,                            // idx 8..11 (MLP1, 2048)
  1024,0,3072,2048,                                   // idx 12..15 (MLP2/3, 1024)
  6144,4096,10240,8192,                               // idx 16..19 (MLP4, 2048)
  14336,12288,18432,16384,                            // idx 20..23 (MLP5, 2048)
  21504,20480};                                       // idx 24..25 (MLP6, 1024)

// ---------------------------------------------------------------------------
// Kernel 1: one block per MLP. Runs the tiny hidden layers (ReLU) and writes
// the last-hidden activations zero-padded to [16][128] f32 into d_ws.
// ---------------------------------------------------------------------------
__global__ void hidden_kernel(Ptrs a, float* __restrict__ ws) {
  const int m   = blockIdx.x;
  const int tid = threadIdx.x;
  __shared__ float buf[2][8][128];

  const int npe  = d_npe[m];
  const int insz = d_insz[m];
  const int h    = d_h[m];
  const int nh   = d_nhid[m];
  const float sow = a.p[0][0];

  // Build input x[p][c]: bits of p (MSB first) then SoW column.
  for (int idx = tid; idx < npe * insz; idx += blockDim.x) {
    int p = idx / insz, c = idx % insz;
    float v = (c == insz - 1) ? sow : (float)((p >> (insz - 2 - c)) & 1);
    buf[0][p][c] = v;
  }
  __syncthreads();

  int cur = 0;
  int fan_in = insz;
  int widx = d_firstW[m];
  for (int l = 0; l < nh; ++l) {
    const float* W = a.p[widx];
    const float* b = a.p[widx + 1];
    const int fan_out = h;
    for (int idx = tid; idx < npe * fan_out; idx += blockDim.x) {
      int p = idx / fan_out, j = idx % fan_out;
      float s = b[j];
      for (int k = 0; k < fan_in; ++k)
        s = fmaf(buf[cur][p][k], W[k * fan_out + j], s);
      buf[1 - cur][p][j] = s > 0.f ? s : 0.f;   // ReLU
    }
    __syncthreads();
    cur = 1 - cur;
    fan_in = fan_out;
    widx += 2;
  }

  // Write [16][128] zero-padded activation tile for the WMMA stage.
  float* dst = ws + m * 2048;
  for (int idx = tid; idx < 16 * 128; idx += blockDim.x) {
    int p = idx >> 7, k = idx & 127;
    dst[idx] = (p < npe && k < h) ? buf[cur][p][k] : 0.f;
  }
}

// ---------------------------------------------------------------------------
// Kernel 2: one wave (32 lanes) per 16-wide output tile of a final layer,
// 8 waves per 256-thread block. D[16x16] = A[16x4] x B[4x16] + C chained
// over K via V_WMMA_F32_16X16X4_F32.
//   A: lane -> M = lane&15, component c -> K = ((lane>=16)?2:0)+c
//   B: lane -> N = lane&15, same K split          (W is [h, out] row-major)
//   D: VGPR v -> M = v + ((lane>=16)?8:0), N = lane&15
// Bias folded in as the initial accumulator (depends only on N).
// K loop: (h>>2) unguarded steps, then ONE wave-uniform guarded tail step
// (act buffer is zero-padded to K=128, so only the W loads need the guard).
// ---------------------------------------------------------------------------
__global__ void wmma_final_kernel(Ptrs a, const float* __restrict__ ws,
                                  float* __restrict__ out) {
  const int tile = blockIdx.x * 8 + (threadIdx.x >> 5);
  int m = 0;
#pragma unroll
  for (int i = 1; i < 7; ++i)
    if (tile >= d_tilecum[i]) m = i;
  const int t  = tile - d_tilecum[m];
  const int j0 = t * 16;

  const int h     = d_h[m];
  const int outsz = d_osz[m];
  const float* __restrict__ act  = ws + m * 2048;
  const float* __restrict__ W    = a.p[d_finalW[m]];
  const float* __restrict__ bias = a.p[d_finalW[m] + 1];

  const int lane  = threadIdx.x & 31;     // 0..31
  const int n     = lane & 15;            // N column (and A-row M for loads)
  const int khalf = (lane >> 4) << 1;     // 0 or 2

  const float bv = bias[j0 + n];
  v8f acc = {bv, bv, bv, bv, bv, bv, bv, bv};

  const float* __restrict__ arow = act + n * 128;     // A: M = lane&15, K padded to 128
  const float* __restrict__ wcol = W + j0 + n;        // B: fixed column, row-strided

  const int fullSteps = h >> 2;           // branch-free main loop
  int kb = khalf;
  for (int s = 0; s < fullSteps; ++s) {
    v2f A, B;
    A.x = arow[kb];
    A.y = arow[kb + 1];
    B.x = wcol[kb * outsz];
    B.y = wcol[(kb + 1) * outsz];
    acc = __builtin_amdgcn_wmma_f32_16x16x4_f32(
        false, A, false, B, (short)0, acc, false, false);
    kb += 4;
  }
  if (h & 3) {                            // wave-uniform tail (h=78 or h=45)
    v2f A, B;
    A.x = arow[kb];                       // zero-padded: always safe
    A.y = arow[kb + 1];
    B.x = (kb     < h) ? wcol[kb * outsz]       : 0.f;
    B.y = (kb + 1 < h) ? wcol[(kb + 1) * outsz] : 0.f;
    acc = __builtin_amdgcn_wmma_f32_16x16x4_f32(
        false, A, false, B, (short)0, acc, false, false);
  }

  const int npe   = d_npe[m];
  const int mbase = (lane >> 4) * 8;      // D-row split across lane halves
  const int sIdx  = d_startIdx[m];
#pragma unroll
  for (int v = 0; v < 8; ++v) {
    const int mm = mbase + v;
    if (mm < npe)
      out[d_outoff[sIdx + mm] + j0 + n] = acc[v];
  }
}

extern "C" void kernel_launch(void* const* d_in, const int* in_sizes, int n_in,
                              void* d_out, int out_size, void* d_ws, size_t ws_size,
                              hipStream_t stream) {
  (void)in_sizes; (void)out_size; (void)ws_size;
  Ptrs a;
  for (int i = 0; i < 35; ++i)
    a.p[i] = (i < n_in) ? (const float*)d_in[i] : nullptr;
  float* ws = (float*)d_ws;   // uses 7 * 2048 * 4 = 56 KB

  hidden_kernel<<<7, 256, 0, stream>>>(a, ws);
  wmma_final_kernel<<<104, 256, 0, stream>>>(a, ws, (float*)d_out);
}